// SparseGrid_90915867721945
// MI455X (gfx1250) — compile-verified
//
#include <hip/hip_runtime.h>

// Sparse voxel grid trilinear interpolation (Plenoxels-style sample kernel).
// Memory-bound random gather: per point, fetch 8 rows of 28 floats and blend.
// Strategy: one thread per point, z-pair-ordered corner fetch (224B contiguous
// pairs -> ~80% cacheline efficiency), float4 (b128) loads/stores, gfx1250
// global_prefetch to start line fills while weight math retires.

#define RESO     192
#define RESO2    (RESO * RESO)
#define DATA_DIM 28   // BASIS_DIM*3 + 4 = 28 floats = 7 x float4 (112 B, 16B aligned)

__global__ __launch_bounds__(256) void SparseGrid_trilerp_kernel(
    const float* __restrict__ points,
    const float* __restrict__ data,
    const int*   __restrict__ links,
    const float* __restrict__ offset,
    const float* __restrict__ scaling,
    float*       __restrict__ out,
    int n)
{
    const int p = blockIdx.x * blockDim.x + threadIdx.x;
    if (p >= n) return;

    const float g   = (float)RESO;
    const float gm1 = g - 1.0f;

    // Wave-uniform parameters -> scalar loads (s_load) via uniformity analysis.
    const float sx = scaling[0] * g;
    const float sy = scaling[1] * g;
    const float sz = scaling[2] * g;
    const float ox = offset[0] * g - 0.5f;
    const float oy = offset[1] * g - 0.5f;
    const float oz = offset[2] * g - 0.5f;

    // Dense across the wave (32 consecutive points = 384 contiguous bytes).
    const float px = points[3 * p + 0];
    const float py = points[3 * p + 1];
    const float pz = points[3 * p + 2];

    float fx = fminf(fmaxf(fmaf(px, sx, ox), 0.0f), gm1);
    float fy = fminf(fmaxf(fmaf(py, sy, oy), 0.0f), gm1);
    float fz = fminf(fmaxf(fmaf(pz, sz, oz), 0.0f), gm1);

    int lx = (int)fx; lx = (lx < RESO - 2) ? lx : (RESO - 2);
    int ly = (int)fy; ly = (ly < RESO - 2) ? ly : (RESO - 2);
    int lz = (int)fz; lz = (lz < RESO - 2) ? lz : (RESO - 2);

    const float bx = fx - (float)lx, ax = 1.0f - bx;
    const float by = fy - (float)ly, ay = 1.0f - by;
    const float bz = fz - (float)lz, az = 1.0f - bz;

    const int base = (lx * RESO + ly) * RESO + lz;

    // 8 corner links fetched as 4 contiguous z-pairs: lk[dx*4 + dy*2 + dz].
    int lk[8];
    #pragma unroll
    for (int c = 0; c < 4; ++c) {
        const int b = base + (c >> 1) * RESO2 + (c & 1) * RESO;
        lk[2 * c + 0] = links[b];
        lk[2 * c + 1] = links[b + 1];
    }

    // Corner weights; empty voxels (link < 0) contribute zero (branchless),
    // index clamped to 0 so the load stays in-bounds.
    const float wx[2] = {ax, bx}, wy[2] = {ay, by}, wz[2] = {az, bz};
    float        w[8];
    const float* rp[8];
    #pragma unroll
    for (int c = 0; c < 8; ++c) {
        const float wc = wx[(c >> 2) & 1] * wy[(c >> 1) & 1] * wz[c & 1];
        w[c]  = (lk[c] >= 0) ? wc : 0.0f;
        const int idx = (lk[c] > 0) ? lk[c] : 0;
        rp[c] = data + (size_t)idx * DATA_DIM;
    }

    // gfx1250 prefetch path: each z-pair is 224 contiguous bytes; prefetch the
    // first two 128B lines of each pair while the weight math above retires.
    #pragma unroll
    for (int c = 0; c < 8; c += 2) {
        __builtin_prefetch((const void*)rp[c], 0, 0);
        __builtin_prefetch((const void*)((const char*)rp[c] + 128), 0, 0);
    }

    float4 acc[7];
    #pragma unroll
    for (int i = 0; i < 7; ++i) acc[i] = make_float4(0.0f, 0.0f, 0.0f, 0.0f);

    // Accumulate two corners (one contiguous z-pair) per block so both hit the
    // same cachelines back-to-back. 7 x b128 loads per corner row.
    #pragma unroll
    for (int c = 0; c < 8; c += 2) {
        const float4* __restrict__ r0 = (const float4*)rp[c];
        const float4* __restrict__ r1 = (const float4*)rp[c + 1];
        const float w0 = w[c], w1 = w[c + 1];
        #pragma unroll
        for (int i = 0; i < 7; ++i) {
            const float4 v0 = r0[i];
            const float4 v1 = r1[i];
            acc[i].x = fmaf(w0, v0.x, fmaf(w1, v1.x, acc[i].x));
            acc[i].y = fmaf(w0, v0.y, fmaf(w1, v1.y, acc[i].y));
            acc[i].z = fmaf(w0, v0.z, fmaf(w1, v1.z, acc[i].z));
            acc[i].w = fmaf(w0, v0.w, fmaf(w1, v1.w, acc[i].w));
        }
    }

    // 112B per thread, dense across the wave (3584 contiguous bytes/wave).
    float4* __restrict__ o = (float4*)(out + (size_t)p * DATA_DIM);
    #pragma unroll
    for (int i = 0; i < 7; ++i) o[i] = acc[i];
}

extern "C" void kernel_launch(void* const* d_in, const int* in_sizes, int n_in,
                              void* d_out, int out_size, void* d_ws, size_t ws_size,
                              hipStream_t stream) {
    const float* points  = (const float*)d_in[0];
    const float* data    = (const float*)d_in[1];
    const int*   links   = (const int*)d_in[2];
    const float* offset  = (const float*)d_in[3];
    const float* scaling = (const float*)d_in[4];
    float*       out     = (float*)d_out;

    const int n     = in_sizes[0] / 3;   // points is (N, 3)
    const int block = 256;               // 8 wave32 waves per workgroup
    const int grid  = (n + block - 1) / block;

    hipLaunchKernelGGL(SparseGrid_trilerp_kernel, dim3(grid), dim3(block), 0, stream,
                       points, data, links, offset, scaling, out, n);
}